// GAT_12532714570149
// MI455X (gfx1250) — compile-verified
//
#include <hip/hip_runtime.h>

#define F_IN   128
#define HID    128
#define HEADS  4
#define OUT_CH 2
#define NEG_SLOPE 0.2f

typedef float v2f __attribute__((ext_vector_type(2)));
typedef float v8f __attribute__((ext_vector_type(8)));

// ---------- helpers ----------
__device__ __forceinline__ unsigned enc_f32(float f) {
    unsigned u = __float_as_uint(f);
    return (u & 0x80000000u) ? ~u : (u | 0x80000000u);
}
__device__ __forceinline__ float dec_f32(unsigned k) {
    return (k & 0x80000000u) ? __uint_as_float(k & 0x7fffffffu)
                             : __uint_as_float(~k);
}
__device__ __forceinline__ float lrelu(float e) { return e > 0.f ? e : NEG_SLOPE * e; }

// ---------- zero fill (float4) ----------
__global__ void fill_zero4(float4* __restrict__ p, long n4) {
    long i = (long)blockIdx.x * blockDim.x + threadIdx.x;
    long stride = (long)gridDim.x * blockDim.x;
    float4 z = {0.f, 0.f, 0.f, 0.f};
    for (; i < n4; i += stride) p[i] = z;
}

// ---------- Layer-1 GEMM: h1[n, o] = sum_k X[n,k] * W1[o,k]  (f32 WMMA 16x16x4) ----------
// grid.x = N/16 row tiles, grid.y = 2, block = 128 (4 waves); each wave does 64 cols.
__global__ __launch_bounds__(128) void gemm1_wmma(const float* __restrict__ X,
                                                  const float* __restrict__ W1,
                                                  float* __restrict__ H) {
    int wave = threadIdx.x >> 5;
    int lane = threadIdx.x & 31;
    int rowTile = blockIdx.x;
    int col0 = (blockIdx.y * 4 + wave) * 64;
    int kh = lane >> 4;                 // lane half selects K pair {0,1} vs {2,3}
    int row = rowTile * 16 + (lane & 15);
    int n0  = col0 + (lane & 15);

    const float* xrow = X + (long)row * F_IN;
    const float* w0 = W1 + (long)(n0 +  0) * F_IN;
    const float* w1 = W1 + (long)(n0 + 16) * F_IN;
    const float* w2 = W1 + (long)(n0 + 32) * F_IN;
    const float* w3 = W1 + (long)(n0 + 48) * F_IN;

    v8f acc0 = {0,0,0,0,0,0,0,0}, acc1 = acc0, acc2 = acc0, acc3 = acc0;

    for (int k0 = 0; k0 < F_IN; k0 += 4) {
        int ko = k0 + 2 * kh;
        v2f a  = *(const v2f*)(xrow + ko);
        v2f b0 = *(const v2f*)(w0 + ko);
        v2f b1 = *(const v2f*)(w1 + ko);
        v2f b2 = *(const v2f*)(w2 + ko);
        v2f b3 = *(const v2f*)(w3 + ko);
        acc0 = __builtin_amdgcn_wmma_f32_16x16x4_f32(false, a, false, b0, (short)0, acc0, false, false);
        acc1 = __builtin_amdgcn_wmma_f32_16x16x4_f32(false, a, false, b1, (short)0, acc1, false, false);
        acc2 = __builtin_amdgcn_wmma_f32_16x16x4_f32(false, a, false, b2, (short)0, acc2, false, false);
        acc3 = __builtin_amdgcn_wmma_f32_16x16x4_f32(false, a, false, b3, (short)0, acc3, false, false);
    }

    // C/D layout: VGPR v -> M = v + 8*laneHalf ; N = lane&15
    int ocol = lane & 15;
    float* hb = H + ((long)(rowTile * 16 + 8 * kh)) * 512 + col0 + ocol;
#pragma unroll
    for (int v = 0; v < 8; ++v) {
        hb[(long)v * 512 +  0] = acc0[v];
        hb[(long)v * 512 + 16] = acc1[v];
        hb[(long)v * 512 + 32] = acc2[v];
        hb[(long)v * 512 + 48] = acc3[v];
    }
}

// ---------- per-node attention halves (layer 1): wave per (node, head) ----------
__global__ __launch_bounds__(256) void alpha1_kernel(const float* __restrict__ H,
                                                     const float* __restrict__ a_src,
                                                     const float* __restrict__ a_dst,
                                                     float* __restrict__ as,
                                                     float* __restrict__ ad, int Nn) {
    long gt = (long)blockIdx.x * blockDim.x + threadIdx.x;
    int gw = (int)(gt >> 5);
    int lane = threadIdx.x & 31;
    int n = gw >> 2, h = gw & 3;
    if (n >= Nn) return;
    const float* hrow = H + (long)n * (HEADS * HID) + h * HID;
    const float* asr = a_src + h * HID;
    const float* adr = a_dst + h * HID;
    float ss = 0.f, sd = 0.f;
#pragma unroll
    for (int j = 0; j < HID / 32; ++j) {
        float v = hrow[lane + 32 * j];
        ss += v * asr[lane + 32 * j];
        sd += v * adr[lane + 32 * j];
    }
#pragma unroll
    for (int o = 16; o > 0; o >>= 1) {
        ss += __shfl_xor(ss, o, 32);
        sd += __shfl_xor(sd, o, 32);
    }
    if (lane == 0) { as[n * HEADS + h] = ss; ad[n * HEADS + h] = sd; }
}

// ---------- edge segment-max (order-preserving uint atomicMax) ----------
__global__ void edge_max(const int* __restrict__ ei, int E_, int Nn,
                         const float* __restrict__ as, const float* __restrict__ ad,
                         unsigned* __restrict__ m) {
    long e = (long)blockIdx.x * blockDim.x + threadIdx.x;
    long tot = (long)E_ + Nn;
    if (e >= tot) return;
    int src = (e < E_) ? ei[e] : (int)(e - E_);
    int dst = (e < E_) ? ei[(long)E_ + e] : (int)(e - E_);
    float4 sv = *(const float4*)(as + (long)src * HEADS);
    float4 dv = *(const float4*)(ad + (long)dst * HEADS);
    unsigned* mb = m + (long)dst * HEADS;
    atomicMax(mb + 0, enc_f32(lrelu(sv.x + dv.x)));
    atomicMax(mb + 1, enc_f32(lrelu(sv.y + dv.y)));
    atomicMax(mb + 2, enc_f32(lrelu(sv.z + dv.z)));
    atomicMax(mb + 3, enc_f32(lrelu(sv.w + dv.w)));
}

// ---------- edge segment-sum of exp(e - max) ----------
__global__ void edge_sum(const int* __restrict__ ei, int E_, int Nn,
                         const float* __restrict__ as, const float* __restrict__ ad,
                         const unsigned* __restrict__ m, float* __restrict__ z) {
    long e = (long)blockIdx.x * blockDim.x + threadIdx.x;
    long tot = (long)E_ + Nn;
    if (e >= tot) return;
    int src = (e < E_) ? ei[e] : (int)(e - E_);
    int dst = (e < E_) ? ei[(long)E_ + e] : (int)(e - E_);
    float4 sv = *(const float4*)(as + (long)src * HEADS);
    float4 dv = *(const float4*)(ad + (long)dst * HEADS);
    uint4  mv = *(const uint4*)(m + (long)dst * HEADS);
    float* zb = z + (long)dst * HEADS;
    atomicAdd(zb + 0, __expf(lrelu(sv.x + dv.x) - dec_f32(mv.x)));
    atomicAdd(zb + 1, __expf(lrelu(sv.y + dv.y) - dec_f32(mv.y)));
    atomicAdd(zb + 2, __expf(lrelu(sv.z + dv.z) - dec_f32(mv.z)));
    atomicAdd(zb + 3, __expf(lrelu(sv.w + dv.w) - dec_f32(mv.w)));
}

// ---------- layer-1 weighted aggregation: wave per edge, 16 ch/lane ----------
__global__ __launch_bounds__(256) void aggregate1(const int* __restrict__ ei, int E_, int Nn,
                                                  const float* __restrict__ H,
                                                  const float* __restrict__ as,
                                                  const float* __restrict__ ad,
                                                  const unsigned* __restrict__ m,
                                                  const float* __restrict__ z,
                                                  float* __restrict__ outp) {
    long gt = (long)blockIdx.x * blockDim.x + threadIdx.x;
    long gw = gt >> 5;
    int lane = threadIdx.x & 31;
    long tot = (long)E_ + Nn;
    if (gw >= tot) return;
    int src = (gw < E_) ? ei[gw] : (int)(gw - E_);
    int dst = (gw < E_) ? ei[(long)E_ + gw] : (int)(gw - E_);
    int h = lane >> 3;                  // 8 lanes per head (128 ch / 16 per lane)
    float ev = lrelu(as[(long)src * HEADS + h] + ad[(long)dst * HEADS + h]);
    float alpha = __expf(ev - dec_f32(m[(long)dst * HEADS + h])) / z[(long)dst * HEADS + h];

    const float4* hs = (const float4*)(H + (long)src * 512 + lane * 16);
    float* ob = outp + (long)dst * 512 + lane * 16;
#pragma unroll
    for (int j = 0; j < 4; ++j) {
        float4 x4 = hs[j];
        atomicAdd(ob + j * 4 + 0, alpha * x4.x);
        atomicAdd(ob + j * 4 + 1, alpha * x4.y);
        atomicAdd(ob + j * 4 + 2, alpha * x4.z);
        atomicAdd(ob + j * 4 + 3, alpha * x4.w);
    }
}

// ---------- bias + ELU in place ----------
__global__ void bias_elu(float* __restrict__ H, const float* __restrict__ b, long total) {
    long i = (long)blockIdx.x * blockDim.x + threadIdx.x;
    if (i >= total) return;
    float v = H[i] + b[i & 511];
    H[i] = v > 0.f ? v : (__expf(v) - 1.f);
}

// ---------- layer-2 projection + attention halves: wave per node ----------
__global__ __launch_bounds__(256) void gemm2_kernel(const float* __restrict__ H,
                                                    const float* __restrict__ W2,
                                                    const float* __restrict__ a_src2,
                                                    const float* __restrict__ a_dst2,
                                                    float* __restrict__ g,
                                                    float* __restrict__ as2,
                                                    float* __restrict__ ad2, int Nn) {
    long gt = (long)blockIdx.x * blockDim.x + threadIdx.x;
    int n = (int)(gt >> 5);
    int lane = threadIdx.x & 31;
    if (n >= Nn) return;
    const float* hrow = H + (long)n * 512;
    float acc[8] = {0.f, 0.f, 0.f, 0.f, 0.f, 0.f, 0.f, 0.f};
    for (int k = lane; k < 512; k += 32) {
        float x = hrow[k];
#pragma unroll
        for (int o = 0; o < 8; ++o) acc[o] += x * W2[o * 512 + k];
    }
#pragma unroll
    for (int o = 0; o < 8; ++o) {
        float v = acc[o];
#pragma unroll
        for (int s = 16; s > 0; s >>= 1) v += __shfl_xor(v, s, 32);
        acc[o] = v;
    }
    if (lane == 0) {
        float* gn = g + (long)n * 8;
#pragma unroll
        for (int o = 0; o < 8; ++o) gn[o] = acc[o];
#pragma unroll
        for (int h = 0; h < HEADS; ++h) {
            as2[(long)n * HEADS + h] = acc[h * 2] * a_src2[h * 2] + acc[h * 2 + 1] * a_src2[h * 2 + 1];
            ad2[(long)n * HEADS + h] = acc[h * 2] * a_dst2[h * 2] + acc[h * 2 + 1] * a_dst2[h * 2 + 1];
        }
    }
}

// ---------- layer-2 aggregation: thread per edge (8 floats) ----------
__global__ void aggregate2(const int* __restrict__ ei, int E_, int Nn,
                           const float* __restrict__ g,
                           const float* __restrict__ as2, const float* __restrict__ ad2,
                           const unsigned* __restrict__ m2, const float* __restrict__ z2,
                           float* __restrict__ out2) {
    long e = (long)blockIdx.x * blockDim.x + threadIdx.x;
    long tot = (long)E_ + Nn;
    if (e >= tot) return;
    int src = (e < E_) ? ei[e] : (int)(e - E_);
    int dst = (e < E_) ? ei[(long)E_ + e] : (int)(e - E_);
    float4 sv = *(const float4*)(as2 + (long)src * HEADS);
    float4 dv = *(const float4*)(ad2 + (long)dst * HEADS);
    uint4  mv = *(const uint4*)(m2 + (long)dst * HEADS);
    float4 zv = *(const float4*)(z2 + (long)dst * HEADS);
    float al[4];
    al[0] = __expf(lrelu(sv.x + dv.x) - dec_f32(mv.x)) / zv.x;
    al[1] = __expf(lrelu(sv.y + dv.y) - dec_f32(mv.y)) / zv.y;
    al[2] = __expf(lrelu(sv.z + dv.z) - dec_f32(mv.z)) / zv.z;
    al[3] = __expf(lrelu(sv.w + dv.w) - dec_f32(mv.w)) / zv.w;
    const float* gs = g + (long)src * 8;
    float* ob = out2 + (long)dst * 8;
#pragma unroll
    for (int h = 0; h < HEADS; ++h) {
        atomicAdd(ob + h * 2 + 0, al[h] * gs[h * 2 + 0]);
        atomicAdd(ob + h * 2 + 1, al[h] * gs[h * 2 + 1]);
    }
}

// ---------- final: mean over heads + bias ----------
__global__ void finalize(const float* __restrict__ out2, const float* __restrict__ b2,
                         float* __restrict__ out, int Nn) {
    long i = (long)blockIdx.x * blockDim.x + threadIdx.x;
    if (i >= (long)Nn * OUT_CH) return;
    int n = (int)(i >> 1), o = (int)(i & 1);
    const float* r = out2 + (long)n * 8;
    float s = r[0 + o] + r[2 + o] + r[4 + o] + r[6 + o];
    out[i] = s * 0.25f + b2[o];
}

extern "C" void kernel_launch(void* const* d_in, const int* in_sizes, int n_in,
                              void* d_out, int out_size, void* d_ws, size_t ws_size,
                              hipStream_t stream) {
    const float* X   = (const float*)d_in[0];
    const int*   EI  = (const int*)d_in[1];
    const float* W1  = (const float*)d_in[2];
    const float* aS1 = (const float*)d_in[3];
    const float* aD1 = (const float*)d_in[4];
    const float* B1  = (const float*)d_in[5];
    const float* W2  = (const float*)d_in[6];
    const float* aS2 = (const float*)d_in[7];
    const float* aD2 = (const float*)d_in[8];
    const float* B2  = (const float*)d_in[9];
    float* OUT = (float*)d_out;

    int Nn = in_sizes[0] / F_IN;   // 50000
    int E_ = in_sizes[1] / 2;      // 800000
    long TOT = (long)E_ + Nn;      // edges + self loops

    // workspace layout (floats); zero-needing buffers contiguous
    float* ws = (float*)d_ws;
    float*    h1   = ws;                                        // Nn*512
    float*    zb   = h1 + (size_t)Nn * 512;                     // zero region start
    float*    out1 = zb;                                        // Nn*512
    unsigned* m1   = (unsigned*)(out1 + (size_t)Nn * 512);      // Nn*4
    float*    z1   = (float*)(m1 + (size_t)Nn * 4);             // Nn*4
    unsigned* m2   = (unsigned*)(z1 + (size_t)Nn * 4);          // Nn*4
    float*    z2   = (float*)(m2 + (size_t)Nn * 4);             // Nn*4
    float*    out2 = z2 + (size_t)Nn * 4;                       // Nn*8
    float*    as1  = out2 + (size_t)Nn * 8;                     // Nn*4
    float*    ad1  = as1 + (size_t)Nn * 4;                      // Nn*4
    float*    g    = ad1 + (size_t)Nn * 4;                      // Nn*8
    float*    as2  = g + (size_t)Nn * 8;                        // Nn*4
    float*    ad2  = as2 + (size_t)Nn * 4;                      // Nn*4

    long zcount = (long)Nn * 536;  // out1 + m1 + z1 + m2 + z2 + out2
    fill_zero4<<<4096, 256, 0, stream>>>((float4*)zb, zcount / 4);

    dim3 g1(Nn / 16, 2);
    gemm1_wmma<<<g1, 128, 0, stream>>>(X, W1, h1);

    long a1t = (long)Nn * HEADS * 32;
    alpha1_kernel<<<(int)((a1t + 255) / 256), 256, 0, stream>>>(h1, aS1, aD1, as1, ad1, Nn);

    int eb = (int)((TOT + 255) / 256);
    edge_max<<<eb, 256, 0, stream>>>(EI, E_, Nn, as1, ad1, m1);
    edge_sum<<<eb, 256, 0, stream>>>(EI, E_, Nn, as1, ad1, m1, z1);

    long aggT = TOT * 32;
    aggregate1<<<(int)((aggT + 255) / 256), 256, 0, stream>>>(EI, E_, Nn, h1, as1, ad1, m1, z1, out1);

    long tot1 = (long)Nn * 512;
    bias_elu<<<(int)((tot1 + 255) / 256), 256, 0, stream>>>(out1, B1, tot1);

    long g2t = (long)Nn * 32;
    gemm2_kernel<<<(int)((g2t + 255) / 256), 256, 0, stream>>>(out1, W2, aS2, aD2, g, as2, ad2, Nn);

    edge_max<<<eb, 256, 0, stream>>>(EI, E_, Nn, as2, ad2, m2);
    edge_sum<<<eb, 256, 0, stream>>>(EI, E_, Nn, as2, ad2, m2, z2);
    aggregate2<<<eb, 256, 0, stream>>>(EI, E_, Nn, g, as2, ad2, m2, z2, out2);

    finalize<<<(int)(((long)Nn * OUT_CH + 255) / 256), 256, 0, stream>>>(out2, B2, OUT, Nn);
}